// Model_37271726195259
// MI455X (gfx1250) — compile-verified
//
#include <hip/hip_runtime.h>

// ---------------------------------------------------------------------------
// 3-layer GraphConv (DGL norm='both', relu) for MI455X (gfx1250, wave32).
// Bottleneck is the width-100 edge scatter-add (L2-resident accumulator).
// The dense (N x 100) @ (100 x 100) GEMM uses V_WMMA_F32_16X16X4_F32.
// ---------------------------------------------------------------------------

typedef __attribute__((ext_vector_type(2))) float v2f;
typedef __attribute__((ext_vector_type(8))) float v8f;

#define DM 100   // hidden width (K and N of the GEMM)
#define SAP 104  // padded K stride for the LDS A tile
#define SBP 112  // padded N stride for the LDS B tile (7 * 16)

// -------------------------------- utility ----------------------------------

__global__ __launch_bounds__(256) void zero_f32(float* __restrict__ p, long n) {
  long i = (long)blockIdx.x * blockDim.x + threadIdx.x;
  long stride = (long)gridDim.x * blockDim.x;
  for (; i < n; i += stride) p[i] = 0.0f;
}

__global__ __launch_bounds__(256) void degree_count(const int* __restrict__ src,
                                                    const int* __restrict__ dst,
                                                    float* __restrict__ cnt_src,
                                                    float* __restrict__ cnt_dst,
                                                    int E) {
  int e = blockIdx.x * blockDim.x + threadIdx.x;
  if (e < E) {
    atomicAdd(&cnt_src[src[e]], 1.0f);
    atomicAdd(&cnt_dst[dst[e]], 1.0f);
  }
}

// in-place counts -> rsqrt(max(count, 1))
__global__ __launch_bounds__(256) void to_rsqrt(float* __restrict__ p, int n) {
  int i = blockIdx.x * blockDim.x + threadIdx.x;
  if (i < n) p[i] = rsqrtf(fmaxf(p[i], 1.0f));
}

// ------------------------------ layer 0 ------------------------------------

// width-2 scatter-add with fused source-degree scaling
__global__ __launch_bounds__(256) void edge_agg_w2(const int* __restrict__ src,
                                                   const int* __restrict__ dst,
                                                   const float* __restrict__ x,
                                                   const float* __restrict__ rs_s,
                                                   float* __restrict__ agg,
                                                   int E) {
  int e = blockIdx.x * blockDim.x + threadIdx.x;
  if (e < E) {
    int s = src[e], d = dst[e];
    float r = rs_s[s];
    atomicAdd(&agg[2 * d + 0], x[2 * s + 0] * r);
    atomicAdd(&agg[2 * d + 1], x[2 * s + 1] * r);
  }
}

// h0s[n][j] = relu((agg2 . W0col_j) * rs_in[n] + b0[j]) * rs_next[n]
__global__ __launch_bounds__(256) void layer0_node(const float* __restrict__ agg,
                                                   const float* __restrict__ W0,
                                                   const float* __restrict__ b0,
                                                   const float* __restrict__ rs_in,
                                                   const float* __restrict__ rs_next,
                                                   float* __restrict__ out, int N) {
  long i = (long)blockIdx.x * blockDim.x + threadIdx.x;
  if (i < (long)N * DM) {
    int n = (int)(i / DM);
    int j = (int)(i - (long)n * DM);
    float z = (agg[2 * n] * W0[j] + agg[2 * n + 1] * W0[DM + j]) * rs_in[n] + b0[j];
    z = fmaxf(z, 0.0f);
    out[i] = z * rs_next[n];
  }
}

// ------------------------------ layer 1 ------------------------------------

// width-100 scatter-add; 25 threads/edge, float4 chunks (rows are 16B aligned)
__global__ __launch_bounds__(256) void edge_agg_w100(const int* __restrict__ sidx,
                                                     const int* __restrict__ didx,
                                                     const float* __restrict__ xs,
                                                     float* __restrict__ agg,
                                                     int total) {
  int i = blockIdx.x * blockDim.x + threadIdx.x;
  if (i < total) {
    unsigned e = (unsigned)i / 25u;
    unsigned ch = (unsigned)i - e * 25u;
    int s = sidx[e], d = didx[e];
    const float4 v = *(const float4*)(xs + (long)s * DM + ch * 4);
    float* a = agg + (long)d * DM + ch * 4;
    atomicAdd(a + 0, v.x);
    atomicAdd(a + 1, v.y);
    atomicAdd(a + 2, v.z);
    atomicAdd(a + 3, v.w);
  }
}

// D = agg(N x 100) @ W(100 x 100) via V_WMMA_F32_16X16X4_F32.
// Block = 7 waves; wave w owns the 16-column tile n0 = 16*w of one 16-row
// M-tile. W and the A tile are staged in LDS (~50 KB of the WGP's 320 KB).
// Epilogue: out = relu(D * rs_in[m] + bias[n]) * rs_next[m].
__global__ __launch_bounds__(224) void gemm100_wmma(const float* __restrict__ agg,
                                                    const float* __restrict__ W,
                                                    const float* __restrict__ bias,
                                                    const float* __restrict__ rs_in,
                                                    const float* __restrict__ rs_next,
                                                    float* __restrict__ out, int N) {
  __shared__ float sB[DM][SBP];  // W[k][n], n zero-padded to 112
  __shared__ float sA[16][SAP];  // A tile, k zero-padded to 104

  const int tid = threadIdx.x;
  const int lane = tid & 31;
  const int wave = tid >> 5;  // 0..6 -> n-tile
  const int m0 = blockIdx.x * 16;

  for (int i = tid; i < DM * SBP; i += 224) {
    int k = i / SBP, n = i - k * SBP;
    sB[k][n] = (n < DM) ? W[k * DM + n] : 0.0f;
  }
  for (int i = tid; i < 16 * SAP; i += 224) {
    int r = i / SAP, k = i - r * SAP;
    int m = m0 + r;
    sA[r][k] = (k < DM && m < N) ? agg[(long)m * DM + k] : 0.0f;
  }
  __syncthreads();

  const int half = lane >> 4;  // 0: K=0,1 / M=r ; 1: K=2,3 / M=r+8
  const int l = lane & 15;
  const int n0 = wave * 16;

  v8f c = {};
#pragma unroll
  for (int kk = 0; kk < DM; kk += 4) {
    v2f a, b;
    a.x = sA[l][kk + 2 * half];          // A 16x4: VGPR0 = K0/K2, VGPR1 = K1/K3
    a.y = sA[l][kk + 2 * half + 1];
    b.x = sB[kk + 2 * half][n0 + l];     // B 4x16: rows striped across lanes
    b.y = sB[kk + 2 * half + 1][n0 + l];
    c = __builtin_amdgcn_wmma_f32_16x16x4_f32(false, a, false, b, (short)0, c,
                                              false, false);
  }

#pragma unroll
  for (int r = 0; r < 8; ++r) {          // C/D: VGPR r -> M=r (lo) / M=r+8 (hi)
    int m = m0 + r + 8 * half;
    int n = n0 + l;
    if (m < N && n < DM) {
      float s = rs_in[m];
      float z = c[r] * s + bias[n];
      z = fmaxf(z, 0.0f);
      out[(long)m * DM + n] = z * rs_next[m];
    }
  }
}

// ------------------------------ layer 2 ------------------------------------

__global__ __launch_bounds__(256) void dot100(const float* __restrict__ h,
                                              const float* __restrict__ w,
                                              float* __restrict__ y, int N) {
  int n = blockIdx.x * blockDim.x + threadIdx.x;
  if (n < N) {
    const float4* hp = (const float4*)(h + (long)n * DM);
    const float4* wp = (const float4*)w;
    float acc = 0.0f;
#pragma unroll
    for (int c = 0; c < DM / 4; ++c) {
      float4 v = hp[c];
      float4 wv = wp[c];
      acc += v.x * wv.x + v.y * wv.y + v.z * wv.z + v.w * wv.w;
    }
    y[n] = acc;
  }
}

__global__ __launch_bounds__(256) void edge_agg_w1(const int* __restrict__ src,
                                                   const int* __restrict__ dst,
                                                   const float* __restrict__ y,
                                                   float* __restrict__ agg, int E) {
  int e = blockIdx.x * blockDim.x + threadIdx.x;
  if (e < E) atomicAdd(&agg[dst[e]], y[src[e]]);
}

__global__ __launch_bounds__(256) void final_node(const float* __restrict__ agg,
                                                  const float* __restrict__ rs_in,
                                                  const float* __restrict__ b2,
                                                  float* __restrict__ out, int N) {
  int n = blockIdx.x * blockDim.x + threadIdx.x;
  if (n < N) out[n] = fmaxf(agg[n] * rs_in[n] + b2[0], 0.0f);
}

// ------------------------------ dispatch -----------------------------------

static inline int cdiv(long a, long b) { return (int)((a + b - 1) / b); }

extern "C" void kernel_launch(void* const* d_in, const int* in_sizes, int n_in,
                              void* d_out, int out_size, void* d_ws, size_t ws_size,
                              hipStream_t stream) {
  const float* x  = (const float*)d_in[0];  // N x 2
  const int*   src = (const int*)d_in[1];   // E
  const int*   dst = (const int*)d_in[2];   // E
  const float* W0 = (const float*)d_in[3];  // 2 x 100
  const float* b0 = (const float*)d_in[4];  // 100
  const float* W1 = (const float*)d_in[5];  // 100 x 100
  const float* b1 = (const float*)d_in[6];  // 100
  const float* W2 = (const float*)d_in[7];  // 100 x 1
  const float* b2 = (const float*)d_in[8];  // 1
  float* out = (float*)d_out;

  const int N = in_sizes[0] / 2;
  const int E = in_sizes[1];

  // Workspace layout (floats): 205*N total (~82 MB)
  float* ws     = (float*)d_ws;
  float* rs_src = ws;                 // N  : counts(src) -> out_deg^-0.5
  float* rs_dst = ws + (long)N;       // N  : counts(dst) -> in_deg^-0.5
  float* agg0   = ws + 2L * N;        // 2N : layer0 accumulator (interleaved)
  float* y2     = agg0;               // N  : reuses agg0 after layer0 is consumed
  float* agg2   = ws + 4L * N;        // N  : layer2 accumulator
  float* big1   = ws + 5L * N;        // 100N : h0 (scaled), later h1 (scaled)
  float* big2   = ws + 105L * N;      // 100N : layer1 accumulator

  // zero: rs_src, rs_dst, agg0, agg2 (contiguous 5N) and the big accumulator
  zero_f32<<<cdiv(5L * N, 256), 256, 0, stream>>>(ws, 5L * N);
  zero_f32<<<cdiv(100L * N, 256), 256, 0, stream>>>(big2, 100L * N);

  // degrees -> rsqrt factors
  degree_count<<<cdiv(E, 256), 256, 0, stream>>>(src, dst, rs_src, rs_dst, E);
  to_rsqrt<<<cdiv(2L * N, 256), 256, 0, stream>>>(rs_src, 2 * N);

  // layer 0: aggregate width-2 (src-scaled), then dense 2->100 + relu,
  // fused with layer-1's source scaling (out_deg' = counts of dst).
  edge_agg_w2<<<cdiv(E, 256), 256, 0, stream>>>(src, dst, x, rs_src, agg0, E);
  layer0_node<<<cdiv((long)N * DM, 256), 256, 0, stream>>>(agg0, W0, b0, rs_dst,
                                                           rs_dst, big1, N);

  // layer 1 (reversed graph): gather big1[dst[e]] -> scatter big2[src[e]]
  edge_agg_w100<<<cdiv((long)E * 25, 256), 256, 0, stream>>>(dst, src, big1, big2,
                                                             E * 25);
  // WMMA GEMM + epilogue (in_deg' = counts of src; layer-2 pre-scale = same)
  gemm100_wmma<<<cdiv(N, 16), 224, 0, stream>>>(big2, W1, b1, rs_src, rs_src,
                                                big1, N);

  // layer 2: shrink to width 1 first, then aggregate
  dot100<<<cdiv(N, 256), 256, 0, stream>>>(big1, W2, y2, N);
  edge_agg_w1<<<cdiv(E, 256), 256, 0, stream>>>(src, dst, y2, agg2, E);
  final_node<<<cdiv(N, 256), 256, 0, stream>>>(agg2, rs_dst, b2, out, N);
}